// SurvivalGNN_47682726920388
// MI455X (gfx1250) — compile-verified
//
#include <hip/hip_runtime.h>

#define N_NODES 50000
#define N_EDGES 800000
#define DIM 128

typedef float v2f __attribute__((ext_vector_type(2)));
typedef float v8f __attribute__((ext_vector_type(8)));

// ---------------------------------------------------------------------------
// Degree / normalization kernels:  deg = 1 (self loop) + scatter count,
// then dinv = rsqrt(deg)  (deg >= 1 always, matches reference's deg>0 branch)
// ---------------------------------------------------------------------------
__global__ void k_init_deg(float* __restrict__ deg) {
    int i = blockIdx.x * blockDim.x + threadIdx.x;
    if (i < N_NODES) deg[i] = 1.0f;
}

__global__ void k_deg_accum(const int* __restrict__ dst, float* __restrict__ deg) {
    int e = blockIdx.x * blockDim.x + threadIdx.x;
    if (e < N_EDGES) atomicAdd(&deg[dst[e]], 1.0f);
}

__global__ void k_deg_to_dinv(float* __restrict__ deg) {
    int i = blockIdx.x * blockDim.x + threadIdx.x;
    if (i < N_NODES) {
        float d = deg[i];
        deg[i] = (d > 0.0f) ? rsqrtf(d) : 0.0f;
    }
}

// ---------------------------------------------------------------------------
// Dense GEMM: out[M x 128] = A[M x 128] @ W[128 x 128] using f32 WMMA.
// One wave computes one 16x16 output tile; 8 waves/block cover 128 columns;
// grid.x = M/16 (M = 50000 is divisible by 16).
//
// V_WMMA_F32_16X16X4_F32 fragment layouts (wave32):
//   A (16x4, M x K):  lane L holds row m = L&15; VGPR0/1 = K = koff, koff+1
//                     where koff = (L>=16) ? 2 : 0.
//   B (4x16, K x N):  lane L holds col n = L&15; VGPR0/1 = K = koff, koff+1.
//   C/D (16x16 f32):  VGPR v: lanes 0-15 -> (M=v,     N=lane),
//                              lanes 16-31 -> (M=v+8, N=lane-16).
// ---------------------------------------------------------------------------
__global__ __launch_bounds__(256)
void k_gemm_wmma_f32(const float* __restrict__ A, const float* __restrict__ W,
                     float* __restrict__ out) {
    const int lane  = threadIdx.x & 31;
    const int wave  = threadIdx.x >> 5;        // 0..7 -> 16-col tile
    const int mTile = blockIdx.x << 4;         // 16 rows per block
    const int n0    = wave << 4;
    const int l15   = lane & 15;
    const int koff  = (lane >> 4) << 1;        // 0 or 2

    const float* arow = A + (size_t)(mTile + l15) * DIM;   // A row, this lane
    const float* bcol = W + n0 + l15;                      // B column, this lane

    v8f acc = {};
#pragma unroll
    for (int kk = 0; kk < DIM; kk += 4) {
        v2f a, b;
        a.x = arow[kk + koff];
        a.y = arow[kk + koff + 1];
        b.x = bcol[(size_t)(kk + koff) * DIM];
        b.y = bcol[(size_t)(kk + koff + 1) * DIM];
        acc = __builtin_amdgcn_wmma_f32_16x16x4_f32(
                  /*neg_a=*/false, a, /*neg_b=*/false, b,
                  /*c_mod=*/(short)0, acc, /*reuse_a=*/false, /*reuse_b=*/false);
    }

    // Store D tile
    const int  rowBase = mTile + ((lane >> 4) << 3);   // +8 for upper half-wave
    float*     orow    = out + (size_t)rowBase * DIM + n0 + l15;
#pragma unroll
    for (int v = 0; v < 8; ++v)
        orow[(size_t)v * DIM] = acc[v];
}

// ---------------------------------------------------------------------------
// agg[i,:] = t[i,:] * dinv[i]^2   (self-loop contribution; also initializes
// the full agg buffer so no memset of poisoned workspace is needed)
// ---------------------------------------------------------------------------
__global__ void k_self_init(const float* __restrict__ t,
                            const float* __restrict__ dinv,
                            float* __restrict__ agg) {
    int idx = blockIdx.x * blockDim.x + threadIdx.x;      // over N*32 float4s
    if (idx >= N_NODES * (DIM / 4)) return;
    int node = idx >> 5;                                  // DIM/4 == 32
    float w  = dinv[node];
    w *= w;
    float4 v = ((const float4*)t)[idx];
    v.x *= w; v.y *= w; v.z *= w; v.w *= w;
    ((float4*)agg)[idx] = v;
}

// ---------------------------------------------------------------------------
// Edge scatter: one wave per edge. Each lane gathers a float4 of the source
// row (L2-resident: node features ~25.6MB << 192MB L2) and atomically adds
// the normalized message into the destination row.
// ---------------------------------------------------------------------------
__global__ __launch_bounds__(256)
void k_scatter(const int* __restrict__ src, const int* __restrict__ dst,
               const float* __restrict__ dinv, const float* __restrict__ t,
               float* __restrict__ agg) {
    const int e = (blockIdx.x << 3) + (threadIdx.x >> 5); // 8 edges / block
    if (e >= N_EDGES) return;
    const int   lane = threadIdx.x & 31;
    const int   s    = src[e];                 // wave-uniform -> scalarized
    const int   d    = dst[e];
    const float nrm  = dinv[s] * dinv[d];

    const float4 v  = ((const float4*)(t + (size_t)s * DIM))[lane];
    float*       ad = agg + (size_t)d * DIM + (lane << 2);
    atomicAdd(ad + 0, v.x * nrm);
    atomicAdd(ad + 1, v.y * nrm);
    atomicAdd(ad + 2, v.z * nrm);
    atomicAdd(ad + 3, v.w * nrm);
}

// ---------------------------------------------------------------------------
// h = relu(agg + b)
// ---------------------------------------------------------------------------
__global__ void k_bias_relu(const float* __restrict__ agg,
                            const float* __restrict__ b,
                            float* __restrict__ h) {
    int idx = blockIdx.x * blockDim.x + threadIdx.x;      // over N*32 float4s
    if (idx >= N_NODES * (DIM / 4)) return;
    int j = idx & (DIM / 4 - 1);
    float4 v  = ((const float4*)agg)[idx];
    float4 bb = ((const float4*)b)[j];
    v.x = fmaxf(v.x + bb.x, 0.0f);
    v.y = fmaxf(v.y + bb.y, 0.0f);
    v.z = fmaxf(v.z + bb.z, 0.0f);
    v.w = fmaxf(v.w + bb.w, 0.0f);
    ((float4*)h)[idx] = v;
}

// ---------------------------------------------------------------------------
// Heads: out[node] = dot(h[node], Wt) + bt ; out[N + node] = dot(h, We) + be
// One wave per node, float4 loads, wave32 shuffle reduction.
// ---------------------------------------------------------------------------
__global__ __launch_bounds__(256)
void k_heads(const float* __restrict__ h,
             const float* __restrict__ Wt, const float* __restrict__ bt,
             const float* __restrict__ We, const float* __restrict__ be,
             float* __restrict__ out) {
    const int node = (blockIdx.x << 3) + (threadIdx.x >> 5);
    if (node >= N_NODES) return;
    const int lane = threadIdx.x & 31;

    const float4 v  = ((const float4*)(h + (size_t)node * DIM))[lane];
    const float4 wt = ((const float4*)Wt)[lane];
    const float4 we = ((const float4*)We)[lane];

    float st = v.x * wt.x + v.y * wt.y + v.z * wt.z + v.w * wt.w;
    float ev = v.x * we.x + v.y * we.y + v.z * we.z + v.w * we.w;

#pragma unroll
    for (int off = 16; off > 0; off >>= 1) {
        st += __shfl_down(st, off, 32);
        ev += __shfl_down(ev, off, 32);
    }
    if (lane == 0) {
        out[node]           = st + bt[0];
        out[N_NODES + node] = ev + be[0];
    }
}

// ---------------------------------------------------------------------------
// Launcher
// ---------------------------------------------------------------------------
extern "C" void kernel_launch(void* const* d_in, const int* in_sizes, int n_in,
                              void* d_out, int out_size, void* d_ws, size_t ws_size,
                              hipStream_t stream) {
    (void)in_sizes; (void)n_in; (void)out_size; (void)ws_size;

    const float* x   = (const float*)d_in[0];
    const int*   ei  = (const int*)  d_in[1];   // [2, E] row-major
    const float* W1  = (const float*)d_in[2];
    const float* b1  = (const float*)d_in[3];
    const float* W2  = (const float*)d_in[4];
    const float* b2  = (const float*)d_in[5];
    const float* Wt  = (const float*)d_in[6];
    const float* bt  = (const float*)d_in[7];
    const float* We  = (const float*)d_in[8];
    const float* be  = (const float*)d_in[9];
    float*       out = (float*)d_out;

    const int* src = ei;            // edge_index[0]
    const int* dst = ei + N_EDGES;  // edge_index[1]

    // Workspace: dinv [N, padded] | bufA [N*128] | bufB [N*128]  (~51.4 MB)
    float* dinv = (float*)d_ws;
    float* bufA = dinv + ((N_NODES + 255) & ~255);
    float* bufB = bufA + (size_t)N_NODES * DIM;

    const int ELEM4   = N_NODES * (DIM / 4);
    const int gNode   = (N_NODES + 255) / 256;
    const int gEdge   = (N_EDGES + 255) / 256;
    const int gElem4  = (ELEM4 + 255) / 256;
    const int gGemm   = N_NODES / 16;          // 3125, exact
    const int gWaveE  = (N_EDGES + 7) / 8;     // 8 edges per 256-thread block
    const int gWaveN  = (N_NODES + 7) / 8;     // 8 nodes per 256-thread block

    // Normalization: deg -> dinv (in place)
    k_init_deg   <<<gNode, 256, 0, stream>>>(dinv);
    k_deg_accum  <<<gEdge, 256, 0, stream>>>(dst, dinv);
    k_deg_to_dinv<<<gNode, 256, 0, stream>>>(dinv);

    // Layer 1: t1 = x @ W1 (bufA); agg1 (bufB); h1 = relu(agg1 + b1) (bufA)
    k_gemm_wmma_f32<<<gGemm,  256, 0, stream>>>(x, W1, bufA);
    k_self_init    <<<gElem4, 256, 0, stream>>>(bufA, dinv, bufB);
    k_scatter      <<<gWaveE, 256, 0, stream>>>(src, dst, dinv, bufA, bufB);
    k_bias_relu    <<<gElem4, 256, 0, stream>>>(bufB, b1, bufA);

    // Layer 2: t2 = h1 @ W2 (bufB); agg2 (bufA); h2 = relu(agg2 + b2) (bufB)
    k_gemm_wmma_f32<<<gGemm,  256, 0, stream>>>(bufA, W2, bufB);
    k_self_init    <<<gElem4, 256, 0, stream>>>(bufB, dinv, bufA);
    k_scatter      <<<gWaveE, 256, 0, stream>>>(src, dst, dinv, bufB, bufA);
    k_bias_relu    <<<gElem4, 256, 0, stream>>>(bufA, b2, bufB);

    // Heads: survival_time | event_status
    k_heads<<<gWaveN, 256, 0, stream>>>(bufB, Wt, bt, We, be, out);
}